// GraphModel_32152125177955
// MI455X (gfx1250) — compile-verified
//
#include <hip/hip_runtime.h>

typedef __attribute__((ext_vector_type(2))) float v2f;
typedef __attribute__((ext_vector_type(8))) float v8f;

// ---------------------------------------------------------------------------
// Degree / normalization kernels
// ---------------------------------------------------------------------------
__global__ void k_init_deg(float* __restrict__ deg, int n) {
  int i = blockIdx.x * blockDim.x + threadIdx.x;
  if (i < n) deg[i] = 1.0f;  // self-loop contributes 1 to every node's degree
}

__global__ void k_count_deg(const long long* __restrict__ dst,
                            float* __restrict__ deg, int E) {
  int e = blockIdx.x * blockDim.x + threadIdx.x;
  if (e < E) atomicAdd(&deg[(unsigned)dst[e]], 1.0f);
}

__global__ void k_dinv(float* __restrict__ deg, int n) {
  int i = blockIdx.x * blockDim.x + threadIdx.x;
  if (i < n) deg[i] = rsqrtf(deg[i]);  // deg >= 1 always
}

// ---------------------------------------------------------------------------
// fp32 WMMA GEMM: one wave (32 lanes) computes one 16x16 tile of
//   D = A[M x K] @ B[K x N]
// via V_WMMA_F32_16X16X4_F32, K stepped by 4, fully unrolled (N, K are
// template constants -> immediate load offsets, zero loop address math).
//
// Epilogue (compile-time flags):
//   SCALE : row-scale by dinv[m] (g = dinv * H) and also store to out2
//           (initializes the scatter accumulator with the self-loop term).
//   BIAS  : add bias[n] (readout layer).
//
// Per-lane fragment layout (CDNA5 ISA 7.12.2, 32-bit, wave32):
//   A 16x4 : lane L -> row M = L%16 ; VGPR0/1 hold K = {0,1} (lanes 0-15)
//            or K = {2,3} (lanes 16-31)
//   B 4x16 : lane L -> col N = L%16 ; VGPR0/1 hold K = {0,1} / {2,3}
//   C/D    : element (vgpr r, lane L) = (M = r + 8*(L>=16), N = L%16)
// ---------------------------------------------------------------------------
template <int N, int K, bool SCALE, bool BIAS>
__global__ void k_wmma_gemm(const float* __restrict__ A,
                            const float* __restrict__ B,
                            const float* __restrict__ dinv,
                            const float* __restrict__ bias,
                            float* __restrict__ out1,
                            float* __restrict__ out2) {
  const unsigned lane = threadIdx.x & 31u;
  const unsigned hi   = lane >> 4;   // 0: lanes 0-15, 1: lanes 16-31
  const unsigned l16  = lane & 15u;

  const unsigned tm = blockIdx.x;    // M/16 tiles
  const unsigned tn = blockIdx.y;    // N/16 tiles

  // Per-lane base pointers; all further accesses use immediate offsets.
  const float* __restrict__ ap = A + (tm * 16u + l16) * (unsigned)K + hi * 2u;
  const float* __restrict__ bp = B + (hi * 2u) * (unsigned)N + tn * 16u + l16;

  v8f c = {};
#pragma unroll
  for (int k0 = 0; k0 < K; k0 += 4) {
    v2f a, b;
    a.x = ap[k0];
    a.y = ap[k0 + 1];
    b.x = bp[k0 * N];
    b.y = bp[k0 * N + N];
    // emits v_wmma_f32_16x16x4_f32
    c = __builtin_amdgcn_wmma_f32_16x16x4_f32(
        /*neg_a=*/false, a, /*neg_b=*/false, b,
        /*c_mod=*/(short)0, c, /*reuse_a=*/false, /*reuse_b=*/false);
  }

  const unsigned ncol = tn * 16u + l16;
  const unsigned mrow0 = tm * 16u + hi * 8u;
#pragma unroll
  for (int r = 0; r < 8; ++r) {
    const unsigned m = mrow0 + r;
    float v = c[r];
    if (SCALE) v *= dinv[m];        // g = dinv * (A @ W)
    if (BIAS)  v += bias[ncol];     // readout bias
    const unsigned o = m * (unsigned)N + ncol;
    out1[o] = v;
    if (SCALE) out2[o] = v;         // acc init == self-loop term g[d]
  }
}

// ---------------------------------------------------------------------------
// Edge scatter: acc[dst] += g[src]   (one wave per edge; F = 64 or 32)
// Coalesced gather of g[src*F .. +F), float atomics into L2-resident acc.
// ---------------------------------------------------------------------------
template <int F>
__global__ void k_scatter(const long long* __restrict__ src,
                          const long long* __restrict__ dst,
                          const float* __restrict__ g,
                          float* __restrict__ acc, int E) {
  const int wid  = (int)((blockIdx.x * blockDim.x + threadIdx.x) >> 5);
  const unsigned lane = threadIdx.x & 31u;
  if (wid >= E) return;
  const unsigned s = (unsigned)src[wid];
  const unsigned d = (unsigned)dst[wid];
  const float* __restrict__ gp = g + s * (unsigned)F + lane;
  float* __restrict__ ap = acc + d * (unsigned)F + lane;
#pragma unroll
  for (int f = 0; f < F; f += 32) atomicAdd(&ap[f], gp[f]);
}

// ---------------------------------------------------------------------------
// Aggregation epilogue: out = relu(dinv[row] * acc + bias[col])
// ---------------------------------------------------------------------------
template <int F>
__global__ void k_epilogue(const float* __restrict__ acc,
                           const float* __restrict__ dinv,
                           const float* __restrict__ bias,
                           float* __restrict__ out, int n) {
  const unsigned i = blockIdx.x * blockDim.x + threadIdx.x;
  if (i >= (unsigned)n * (unsigned)F) return;
  const unsigned m = i / (unsigned)F;   // F = power of two -> shift
  const unsigned f = i & (unsigned)(F - 1);
  float v = dinv[m] * acc[i] + bias[f];
  out[i] = fmaxf(v, 0.0f);
}

// ---------------------------------------------------------------------------
// Driver
// ---------------------------------------------------------------------------
extern "C" void kernel_launch(void* const* d_in, const int* in_sizes, int n_in,
                              void* d_out, int out_size, void* d_ws, size_t ws_size,
                              hipStream_t stream) {
  const float*     x  = (const float*)d_in[0];        // [n, 128]
  const long long* ei = (const long long*)d_in[1];    // [2, E] int64
  const float*     W1 = (const float*)d_in[2];        // [128, 64]
  const float*     b1 = (const float*)d_in[3];        // [64]
  const float*     W2 = (const float*)d_in[4];        // [64, 32]
  const float*     b2 = (const float*)d_in[5];        // [32]
  const float*     Wl = (const float*)d_in[6];        // [32, 128]
  const float*     bl = (const float*)d_in[7];        // [128]

  const int n = in_sizes[0] / 128;   // 50000 (multiple of 16: 3125 tiles)
  const int E = in_sizes[1] / 2;     // 800000
  const long long* src = ei;
  const long long* dst = ei + E;

  // Workspace layout (fp32): total ~ n*(1 + 2*64 + 2*32)*4B ~= 38.6 MB
  float* ws   = (float*)d_ws;
  float* dinv = ws;                         // n
  float* g1   = dinv + n;                   // n*64  (also reused for A1)
  float* acc1 = g1   + (size_t)n * 64;      // n*64
  float* g2   = acc1 + (size_t)n * 64;      // n*32  (also reused for A2)
  float* acc2 = g2   + (size_t)n * 32;      // n*32
  float* out  = (float*)d_out;              // n*128

  const int T = 256;

  // --- degrees & symmetric normalization -----------------------------------
  k_init_deg <<<(n + T - 1) / T, T, 0, stream>>>(dinv, n);
  k_count_deg<<<(E + T - 1) / T, T, 0, stream>>>(dst, dinv, E);
  k_dinv     <<<(n + T - 1) / T, T, 0, stream>>>(dinv, n);

  // --- layer 1: g1 = dinv * (X @ W1); acc1 = g1 (self-loop) ----------------
  {
    dim3 grid(n / 16, 64 / 16);
    k_wmma_gemm<64, 128, true, false>
        <<<grid, 32, 0, stream>>>(x, W1, dinv, nullptr, g1, acc1);
  }
  k_scatter<64> <<<(E + 7) / 8, T, 0, stream>>>(src, dst, g1, acc1, E);
  k_epilogue<64><<<((size_t)n * 64 + T - 1) / T, T, 0, stream>>>(acc1, dinv,
                                                                 b1, g1, n);

  // --- layer 2: g2 = dinv * (A1 @ W2); acc2 = g2 ---------------------------
  {
    dim3 grid(n / 16, 32 / 16);
    k_wmma_gemm<32, 64, true, false>
        <<<grid, 32, 0, stream>>>(g1, W2, dinv, nullptr, g2, acc2);
  }
  k_scatter<32> <<<(E + 7) / 8, T, 0, stream>>>(src, dst, g2, acc2, E);
  k_epilogue<32><<<((size_t)n * 32 + T - 1) / T, T, 0, stream>>>(acc2, dinv,
                                                                 b2, g2, n);

  // --- readout: out = A2 @ Wl + bl -----------------------------------------
  {
    dim3 grid(n / 16, 128 / 16);
    k_wmma_gemm<128, 32, false, true>
        <<<grid, 32, 0, stream>>>(g2, Wl, nullptr, bl, out, nullptr);
  }
}